// RNNAE_14791867367996
// MI455X (gfx1250) — compile-verified
//
#include <hip/hip_runtime.h>

// ---------------------------------------------------------------------------
// RNN autoencoder for MI455X (gfx1250, wave32).
//   proj_enc / proj_dec : parallel skinny GEMMs via V_WMMA_F32_16X16X4_F32
//   scan_enc / scan_dec : single-wave serial LSTM scans. One lane per gate;
//                         v_readlane gathers gate values into uniform scalars
//                         so every lane redundantly computes the full cell
//                         update (no LDS-crossbar latency on critical path).
//                         Activations via v_tanh_f32 (CDNA5) when available.
//   head                : parallel fused MLP epilogue
// ---------------------------------------------------------------------------

typedef __attribute__((ext_vector_type(2))) float v2f;
typedef __attribute__((ext_vector_type(8))) float v8f;

#define T_LEN 524288
#define HE 3
#define HD 5
#define GE 12   // 4*HE
#define GD 20   // 4*HD
#define PAD_ROWS 64

__device__ __forceinline__ float read_lane_f(float v, int lane) {
    return __int_as_float(__builtin_amdgcn_readlane(__float_as_int(v), lane));
}

__device__ __forceinline__ float fast_tanh(float x) {
#if __has_builtin(__builtin_amdgcn_tanhf)
    return __builtin_amdgcn_tanhf(x);          // v_tanh_f32 (gfx1250)
#else
    // tanh(x) = 2/(1+exp2(-2x*log2e)) - 1  -> v_exp_f32 + v_rcp_f32
    float e = __builtin_amdgcn_exp2f(-2.88539008177792681472f * x);
    return fmaf(2.0f, __builtin_amdgcn_rcpf(1.0f + e), -1.0f);
#endif
}

// ---------------------------------------------------------------------------
// Encoder input projection: Xg[t, 0:12] = x[t, 0:8] @ Wih_e.T + (bih_e+bhh_e)
// A tile: 16x8 (two K=4 chunks), B: Wih_e.T (8x12, N padded to 16).
// ---------------------------------------------------------------------------
__global__ void proj_enc_kernel(const float* __restrict__ x,
                                const float* __restrict__ Wih,   // [12,8]
                                const float* __restrict__ bih,
                                const float* __restrict__ bhh,
                                float* __restrict__ Xg) {
    const int lane   = threadIdx.x & 31;
    const int half   = lane >> 4;       // 0: K=+0/+1, 1: K=+2/+3
    const int lp     = lane & 15;
    const int wave   = blockIdx.x * (blockDim.x >> 5) + (threadIdx.x >> 5);
    const int nwaves = gridDim.x * (blockDim.x >> 5);
    const int ntiles = T_LEN / 16;
    const int k0a = half ? 2 : 0, k0b = half ? 3 : 1;

    // B = Wih^T : B[k][n] = Wih[n*8 + k], n = lp (valid n < 12)
    v2f b0, b1;
    {
        const bool nv = (lp < GE);
        b0.x = nv ? Wih[lp * 8 + 0 + k0a] : 0.0f;
        b0.y = nv ? Wih[lp * 8 + 0 + k0b] : 0.0f;
        b1.x = nv ? Wih[lp * 8 + 4 + k0a] : 0.0f;
        b1.y = nv ? Wih[lp * 8 + 4 + k0b] : 0.0f;
    }
    const float bias = (lp < GE) ? (bih[lp] + bhh[lp]) : 0.0f;

    for (int tile = wave; tile < ntiles; tile += nwaves) {
        const int row = (tile << 4) + lp;   // both halves read same 16 rows
        v2f a0, a1;
        a0.x = x[row * 8 + 0 + k0a];
        a0.y = x[row * 8 + 0 + k0b];
        a1.x = x[row * 8 + 4 + k0a];
        a1.y = x[row * 8 + 4 + k0b];

        v8f acc;
        #pragma unroll
        for (int r = 0; r < 8; ++r) acc[r] = bias;
        acc = __builtin_amdgcn_wmma_f32_16x16x4_f32(false, a0, false, b0,
                                                    (short)0, acc, false, false);
        acc = __builtin_amdgcn_wmma_f32_16x16x4_f32(false, a1, false, b1,
                                                    (short)0, acc, false, false);

        if (lp < GE) {
            const int mbase = (tile << 4) + (half << 3);
            #pragma unroll
            for (int r = 0; r < 8; ++r)
                Xg[(size_t)(mbase + r) * GE + lp] = acc[r];
        }
    }
}

// ---------------------------------------------------------------------------
// Decoder input projection: Sg[t, 0:20] = s[t, 0:6] @ Wih_d[:, 0:6].T + bias
// K padded 6->8 (two K=4 chunks), N = 20 -> two N-tiles (16 + 4).
// ---------------------------------------------------------------------------
__global__ void proj_dec_kernel(const float* __restrict__ s,
                                const float* __restrict__ Wih,   // [20,7]
                                const float* __restrict__ bih,
                                const float* __restrict__ bhh,
                                float* __restrict__ Sg) {
    const int lane   = threadIdx.x & 31;
    const int half   = lane >> 4;
    const int lp     = lane & 15;
    const int wave   = blockIdx.x * (blockDim.x >> 5) + (threadIdx.x >> 5);
    const int nwaves = gridDim.x * (blockDim.x >> 5);
    const int ntiles = T_LEN / 16;
    const int k0a = half ? 2 : 0, k0b = half ? 3 : 1;

    // B[k][n] = Wih[n*7 + k], k in 0..5, n in 0..19
    v2f b00, b01, b10, b11;   // b<NTile><KChunk>
    {
        const int n0 = lp;                       // always < 20
        b00.x = Wih[n0 * 7 + k0a];
        b00.y = Wih[n0 * 7 + k0b];
        b01.x = (4 + k0a < 6) ? Wih[n0 * 7 + 4 + k0a] : 0.0f;
        b01.y = (4 + k0b < 6) ? Wih[n0 * 7 + 4 + k0b] : 0.0f;
        const int n1 = 16 + lp;
        const bool nv = (n1 < GD);
        b10.x = nv ? Wih[n1 * 7 + k0a] : 0.0f;
        b10.y = nv ? Wih[n1 * 7 + k0b] : 0.0f;
        b11.x = (nv && (4 + k0a < 6)) ? Wih[n1 * 7 + 4 + k0a] : 0.0f;
        b11.y = (nv && (4 + k0b < 6)) ? Wih[n1 * 7 + 4 + k0b] : 0.0f;
    }
    const float bias0 = bih[lp] + bhh[lp];
    const float bias1 = (16 + lp < GD) ? (bih[16 + lp] + bhh[16 + lp]) : 0.0f;

    for (int tile = wave; tile < ntiles; tile += nwaves) {
        const int row = (tile << 4) + lp;
        v2f a0, a1;
        a0.x = s[row * 6 + k0a];
        a0.y = s[row * 6 + k0b];
        a1.x = (4 + k0a < 6) ? s[row * 6 + 4 + k0a] : 0.0f;
        a1.y = (4 + k0b < 6) ? s[row * 6 + 4 + k0b] : 0.0f;

        v8f acc0, acc1;
        #pragma unroll
        for (int r = 0; r < 8; ++r) { acc0[r] = bias0; acc1[r] = bias1; }
        acc0 = __builtin_amdgcn_wmma_f32_16x16x4_f32(false, a0, false, b00,
                                                     (short)0, acc0, false, false);
        acc0 = __builtin_amdgcn_wmma_f32_16x16x4_f32(false, a1, false, b01,
                                                     (short)0, acc0, false, false);
        acc1 = __builtin_amdgcn_wmma_f32_16x16x4_f32(false, a0, false, b10,
                                                     (short)0, acc1, false, false);
        acc1 = __builtin_amdgcn_wmma_f32_16x16x4_f32(false, a1, false, b11,
                                                     (short)0, acc1, false, false);

        const int mbase = (tile << 4) + (half << 3);
        #pragma unroll
        for (int r = 0; r < 8; ++r)
            Sg[(size_t)(mbase + r) * GD + lp] = acc0[r];
        if (lp < GD - 16) {
            #pragma unroll
            for (int r = 0; r < 8; ++r)
                Sg[(size_t)(mbase + r) * GD + 16 + lp] = acc1[r];
        }
    }
}

// ---------------------------------------------------------------------------
// Encoder LSTM scan: 1 wave; lane j<12 owns gate j. Gate values are pulled
// into uniform scalars with v_readlane; every lane keeps the full (h, c)
// state, so there is no cross-lane broadcast on the critical path.
// ---------------------------------------------------------------------------
__global__ void scan_enc_kernel(const float* __restrict__ Xg,    // [T,12]+pad
                                const float* __restrict__ Whh,   // [12,3]
                                const float* __restrict__ h0v,
                                const float* __restrict__ c0v,
                                const float* __restrict__ Wfce,  // [1,3]
                                const float* __restrict__ bfce,
                                float* __restrict__ zout) {
    const int j  = threadIdx.x;
    const int jj = (j < GE) ? j : 0;
    const float w0 = Whh[jj * HE + 0];
    const float w1 = Whh[jj * HE + 1];
    const float w2 = Whh[jj * HE + 2];
    const bool  isG = (jj >= 2 * HE) && (jj < 3 * HE);   // tanh gate
    // sigmoid(x) = 0.5*tanh(0.5x)+0.5 ; tanh gate: plain tanh
    const float sc = isG ? 1.0f : 0.5f;
    const float mm = isG ? 1.0f : 0.5f;
    const float aa = isG ? 0.0f : 0.5f;

    float h0 = h0v[0], h1 = h0v[1], h2 = h0v[2];
    float c0 = c0v[0], c1 = c0v[1], c2 = c0v[2];
    float cur = Xg[jj];

    for (int t = 0; t < T_LEN; ++t) {
        __builtin_prefetch(Xg + (size_t)(t + 63) * GE + jj, 0, 0);
        float nxt = Xg[(size_t)(t + 1) * GE + jj];
        float g   = fmaf(w0, h0, w1 * h1) + fmaf(w2, h2, cur);
        float val = fmaf(mm, fast_tanh(sc * g), aa);
        float i0 = read_lane_f(val, 0), i1 = read_lane_f(val, 1), i2 = read_lane_f(val, 2);
        float f0 = read_lane_f(val, 3), f1 = read_lane_f(val, 4), f2 = read_lane_f(val, 5);
        float g0 = read_lane_f(val, 6), g1 = read_lane_f(val, 7), g2 = read_lane_f(val, 8);
        float o0 = read_lane_f(val, 9), o1 = read_lane_f(val, 10), o2 = read_lane_f(val, 11);
        c0 = fmaf(f0, c0, i0 * g0);
        c1 = fmaf(f1, c1, i1 * g1);
        c2 = fmaf(f2, c2, i2 * g2);
        h0 = o0 * fast_tanh(c0);
        h1 = o1 * fast_tanh(c1);
        h2 = o2 * fast_tanh(c2);
        cur = nxt;
    }
    if (j == 0) {
        zout[0] = fmaf(h0, Wfce[0], fmaf(h1, Wfce[1], fmaf(h2, Wfce[2], bfce[0])));
    }
}

// ---------------------------------------------------------------------------
// Decoder LSTM scan: lane j<20 owns gate j; z folded in as per-gate constant.
// ---------------------------------------------------------------------------
__global__ void scan_dec_kernel(const float* __restrict__ Sg,    // [T,20]+pad
                                const float* __restrict__ Whh,   // [20,5]
                                const float* __restrict__ Wih,   // [20,7]
                                const float* __restrict__ h0v,
                                const float* __restrict__ c0v,
                                const float* __restrict__ zin,
                                float* __restrict__ hs) {
    const int j  = threadIdx.x;
    const int jj = (j < GD) ? j : 0;
    const float w0 = Whh[jj * HD + 0];
    const float w1 = Whh[jj * HD + 1];
    const float w2 = Whh[jj * HD + 2];
    const float w3 = Whh[jj * HD + 3];
    const float w4 = Whh[jj * HD + 4];
    const bool  isG = (jj >= 2 * HD) && (jj < 3 * HD);
    const float sc = isG ? 1.0f : 0.5f;
    const float mm = isG ? 1.0f : 0.5f;
    const float aa = isG ? 0.0f : 0.5f;
    const float z  = zin[0];
    const float zc = z * Wih[jj * 7 + 6];       // z contribution: constant/step

    float h0 = h0v[0], h1 = h0v[1], h2 = h0v[2], h3 = h0v[3], h4 = h0v[4];
    float c0 = c0v[0], c1 = c0v[1], c2 = c0v[2], c3 = c0v[3], c4 = c0v[4];
    float cur = Sg[jj] + zc;

    for (int t = 0; t < T_LEN; ++t) {
        __builtin_prefetch(Sg + (size_t)(t + 63) * GD + jj, 0, 0);
        float nxt = Sg[(size_t)(t + 1) * GD + jj];
        // tree-reduced recurrent dot (depth ~3)
        float g = (fmaf(w0, h0, w1 * h1) + fmaf(w2, h2, w3 * h3)) + fmaf(w4, h4, cur);
        float val = fmaf(mm, fast_tanh(sc * g), aa);
        float i0 = read_lane_f(val, 0),  i1 = read_lane_f(val, 1);
        float i2 = read_lane_f(val, 2),  i3 = read_lane_f(val, 3),  i4 = read_lane_f(val, 4);
        float f0 = read_lane_f(val, 5),  f1 = read_lane_f(val, 6);
        float f2 = read_lane_f(val, 7),  f3 = read_lane_f(val, 8),  f4 = read_lane_f(val, 9);
        float g0 = read_lane_f(val, 10), g1 = read_lane_f(val, 11);
        float g2 = read_lane_f(val, 12), g3 = read_lane_f(val, 13), g4 = read_lane_f(val, 14);
        float o0 = read_lane_f(val, 15), o1 = read_lane_f(val, 16);
        float o2 = read_lane_f(val, 17), o3 = read_lane_f(val, 18), o4 = read_lane_f(val, 19);
        c0 = fmaf(f0, c0, i0 * g0);
        c1 = fmaf(f1, c1, i1 * g1);
        c2 = fmaf(f2, c2, i2 * g2);
        c3 = fmaf(f3, c3, i3 * g3);
        c4 = fmaf(f4, c4, i4 * g4);
        h0 = o0 * fast_tanh(c0);
        h1 = o1 * fast_tanh(c1);
        h2 = o2 * fast_tanh(c2);
        h3 = o3 * fast_tanh(c3);
        h4 = o4 * fast_tanh(c4);
        // store hs[t, :] from lanes 0..4 (h* are uniform; select by lane)
        float sv = h0;
        sv = (j == 1) ? h1 : sv;
        sv = (j == 2) ? h2 : sv;
        sv = (j == 3) ? h3 : sv;
        sv = (j == 4) ? h4 : sv;
        if (j < HD) hs[(size_t)t * HD + j] = sv;
        cur = nxt + zc;
    }
}

// ---------------------------------------------------------------------------
// Head: a_hat = relu(relu(hs@W1.T+b1)@W2.T+b2)@W3.T+b3 ; one row per thread.
// ---------------------------------------------------------------------------
__global__ void head_kernel(const float* __restrict__ hs,
                            const float* __restrict__ W1, const float* __restrict__ b1,
                            const float* __restrict__ W2, const float* __restrict__ b2,
                            const float* __restrict__ W3, const float* __restrict__ b3,
                            float* __restrict__ out) {
    const int t = blockIdx.x * blockDim.x + threadIdx.x;
    if (t >= T_LEN) return;
    float h[HD];
    #pragma unroll
    for (int i = 0; i < HD; ++i) h[i] = hs[(size_t)t * HD + i];
    float y1[2 * HD];
    #pragma unroll
    for (int o = 0; o < 2 * HD; ++o) {
        float a = b1[o];
        #pragma unroll
        for (int i = 0; i < HD; ++i) a = fmaf(W1[o * HD + i], h[i], a);
        y1[o] = fmaxf(a, 0.0f);
    }
    float y2[2 * HD];
    #pragma unroll
    for (int o = 0; o < 2 * HD; ++o) {
        float a = b2[o];
        #pragma unroll
        for (int i = 0; i < 2 * HD; ++i) a = fmaf(W2[o * 2 * HD + i], y1[i], a);
        y2[o] = fmaxf(a, 0.0f);
    }
    #pragma unroll
    for (int oc = 0; oc < 2; ++oc) {
        float a = b3[oc];
        #pragma unroll
        for (int i = 0; i < 2 * HD; ++i) a = fmaf(W3[oc * 2 * HD + i], y2[i], a);
        out[(size_t)t * 2 + oc] = a;
    }
}

// ---------------------------------------------------------------------------
extern "C" void kernel_launch(void* const* d_in, const int* in_sizes, int n_in,
                              void* d_out, int out_size, void* d_ws, size_t ws_size,
                              hipStream_t stream) {
    (void)in_sizes; (void)n_in; (void)out_size; (void)ws_size;
    const float* x      = (const float*)d_in[0];
    const float* s      = (const float*)d_in[1];
    const float* h0_enc = (const float*)d_in[2];
    const float* c0_enc = (const float*)d_in[3];
    const float* h0_dec = (const float*)d_in[4];
    const float* c0_dec = (const float*)d_in[5];
    const float* Wih_e  = (const float*)d_in[6];
    const float* Whh_e  = (const float*)d_in[7];
    const float* bih_e  = (const float*)d_in[8];
    const float* bhh_e  = (const float*)d_in[9];
    const float* Wfce   = (const float*)d_in[10];
    const float* bfce   = (const float*)d_in[11];
    const float* Wih_d  = (const float*)d_in[12];
    const float* Whh_d  = (const float*)d_in[13];
    const float* bih_d  = (const float*)d_in[14];
    const float* bhh_d  = (const float*)d_in[15];
    const float* W1     = (const float*)d_in[16];
    const float* b1     = (const float*)d_in[17];
    const float* W2     = (const float*)d_in[18];
    const float* b2     = (const float*)d_in[19];
    const float* W3     = (const float*)d_in[20];
    const float* b3     = (const float*)d_in[21];
    float* out = (float*)d_out;

    // workspace layout (floats), padded so next-row loads / prefetch stay
    // inside the allocation
    float* ws = (float*)d_ws;
    float* Xg = ws;                                           // T*12 (+pad)
    float* Sg = Xg + (size_t)T_LEN * GE + PAD_ROWS * GE;      // T*20 (+pad)
    float* hs = Sg + (size_t)T_LEN * GD + PAD_ROWS * GD;      // T*5
    float* z  = hs + (size_t)T_LEN * HD;                      // 1

    proj_enc_kernel<<<dim3(256), dim3(256), 0, stream>>>(x, Wih_e, bih_e, bhh_e, Xg);
    proj_dec_kernel<<<dim3(256), dim3(256), 0, stream>>>(s, Wih_d, bih_d, bhh_d, Sg);
    scan_enc_kernel<<<dim3(1), dim3(32), 0, stream>>>(Xg, Whh_e, h0_enc, c0_enc, Wfce, bfce, z);
    scan_dec_kernel<<<dim3(1), dim3(32), 0, stream>>>(Sg, Whh_d, Wih_d, h0_dec, c0_dec, z, hs);
    head_kernel<<<dim3(T_LEN / 256), dim3(256), 0, stream>>>(hs, W1, b1, W2, b2, W3, b3, out);
}